// FastAttention_89876485636714
// MI455X (gfx1250) — compile-verified
//
#include <hip/hip_runtime.h>
#include <math.h>

typedef float v2f __attribute__((ext_vector_type(2)));
typedef float v8f __attribute__((ext_vector_type(8)));

// D = A(16x4,f32) * B(4x16,f32) + C(16x16,f32)
#define WMMA_F32(a, b, c) \
  __builtin_amdgcn_wmma_f32_16x16x4_f32(false, (a), false, (b), (short)0, (c), false, false)

constexpr int B_ = 2, H_ = 16, N_ = 4096, D_ = 64, F_ = 256;
constexpr int BH = B_ * H_;          // 32
constexpr int SPLITS = 8;            // K-splits per (b,h) for kv partials
constexpr int ROWS1 = N_ / SPLITS;   // 512 rows per pass-1 block
constexpr int CH1 = ROWS1 / 16;      // 32 chunks of 16 rows
constexpr int KP_PITCH = 260;        // 256-feature rows, padded (bank stride 4)
constexpr int KQ_PITCH = 68;         // 64-col rows, padded (bank stride 4)

// ---------------------------------------------------------------------------
// Pass 1: k -> k' (on the fly, in LDS) -> kv partial [F x D] and ksum partial.
// grid = BH * SPLITS blocks, 512 threads (16 waves).
// ---------------------------------------------------------------------------
__global__ __launch_bounds__(512)
void FastAttention_kv_kernel(const float* __restrict__ k,
                             const float* __restrict__ v,
                             const float* __restrict__ P,
                             float* __restrict__ kvp,
                             float* __restrict__ ksump)
{
  __shared__ float kLDS[16 * KQ_PITCH];    // 16 rows x 64 (padded)
  __shared__ float kpLDS[16 * KP_PITCH];   // k' tile: 16 rows x 256 (padded)
  __shared__ float maxbuf[16 * 16];        // [row][wave]
  __shared__ float rowmax[16];

  const int tid  = threadIdx.x;
  const int lane = tid & 31;
  const int wave = tid >> 5;
  const int l16  = lane & 15;
  const int hi   = lane >> 4;              // 0 or 1 (lane half)
  const int bh    = blockIdx.x / SPLITS;
  const int split = blockIdx.x % SPLITS;
  const int n0    = split * ROWS1;
  const float* kb = k + (size_t)bh * N_ * D_;
  const float* vb = v + (size_t)bh * N_ * D_;
  const int f0 = wave * 16;                // this wave's feature block

  // Preload projection-matrix B-fragments for this wave's 16 features.
  v2f pB[16];
#pragma unroll
  for (int kk = 0; kk < 16; ++kk) {
    int d = kk * 4 + 2 * hi;
    pB[kk].x = P[(size_t)d * F_ + f0 + l16];
    pB[kk].y = P[(size_t)(d + 1) * F_ + f0 + l16];
  }

  v8f kvacc[4] = {};   // 4 e-tiles (16x16 each) for feature rows f0..f0+15
  float ksacc = 0.0f;

  for (int c = 0; c < CH1; ++c) {
    __syncthreads();
    // Stage 16x64 k tile (coalesced).
    for (int i = tid; i < 16 * 64; i += 512)
      kLDS[(i >> 6) * KQ_PITCH + (i & 63)] =
          kb[(size_t)(n0 + c * 16 + (i >> 6)) * D_ + (i & 63)];
    __syncthreads();

    // Projection: rows x features tile, K = D = 64 (16 WMMA ops).
    v8f acc = {};
#pragma unroll
    for (int kk = 0; kk < 16; ++kk) {
      v2f a = *(const v2f*)&kLDS[l16 * KQ_PITCH + kk * 4 + 2 * hi];
      acc = WMMA_F32(a, pB[kk], acc);
    }

    // Per-row max across this wave's 16 features (16-lane butterfly).
#pragma unroll
    for (int r = 0; r < 8; ++r) {
      float m = acc[r];
      m = fmaxf(m, __shfl_xor(m, 1, 32));
      m = fmaxf(m, __shfl_xor(m, 2, 32));
      m = fmaxf(m, __shfl_xor(m, 4, 32));
      m = fmaxf(m, __shfl_xor(m, 8, 32));
      if (l16 == 0) maxbuf[(r + hi * 8) * 16 + wave] = m;
    }
    __syncthreads();
    if (tid < 16) {
      float m = maxbuf[tid * 16];
#pragma unroll
      for (int w = 1; w < 16; ++w) m = fmaxf(m, maxbuf[tid * 16 + w]);
      rowmax[tid] = m;
    }
    __syncthreads();

    // exp(x - rowmax)/sqrt(F); stash k' tile; accumulate ksum.
#pragma unroll
    for (int r = 0; r < 8; ++r) {
      float e = __expf(acc[r] - rowmax[r + hi * 8]) * 0.0625f;
      ksacc += e;
      kpLDS[(r + hi * 8) * KP_PITCH + f0 + l16] = e;
    }
    __syncthreads();

    // kv += k'^T (16f x 16n) * v (16n x 16e): 4 WMMA per e-tile.
#pragma unroll
    for (int kk = 0; kk < 4; ++kk) {
      int nl = kk * 4 + 2 * hi;
      v2f a;
      a.x = kpLDS[nl * KP_PITCH + f0 + l16];
      a.y = kpLDS[(nl + 1) * KP_PITCH + f0 + l16];
      size_t vrow = (size_t)(n0 + c * 16 + nl);
#pragma unroll
      for (int e = 0; e < 4; ++e) {
        v2f b;
        b.x = vb[vrow * D_ + e * 16 + l16];
        b.y = vb[(vrow + 1) * D_ + e * 16 + l16];
        kvacc[e] = WMMA_F32(a, b, kvacc[e]);
      }
    }
  }

  // Write kv partial [F x D] for this (bh, split).
  float* kvo = kvp + ((size_t)bh * SPLITS + split) * F_ * D_;
#pragma unroll
  for (int e = 0; e < 4; ++e)
#pragma unroll
    for (int r = 0; r < 8; ++r)
      kvo[(size_t)(f0 + r + hi * 8) * D_ + e * 16 + l16] = kvacc[e][r];

  // ksum partial: combine lane halves (distinct rows, same feature).
  ksacc += __shfl_xor(ksacc, 16, 32);
  if (hi == 0)
    ksump[((size_t)bh * SPLITS + split) * F_ + f0 + l16] = ksacc;
}

// ---------------------------------------------------------------------------
// Pass 1b: fixed-order reduction of kv / ksum partials (deterministic).
// ---------------------------------------------------------------------------
__global__ __launch_bounds__(256)
void FastAttention_reduce_kernel(const float* __restrict__ kvp,
                                 const float* __restrict__ ksump,
                                 float* __restrict__ kv,
                                 float* __restrict__ ksum)
{
  int i = blockIdx.x * 256 + threadIdx.x;
  if (i < BH * F_ * D_) {
    int bh = i / (F_ * D_);
    int fe = i % (F_ * D_);
    float s = 0.0f;
#pragma unroll
    for (int p = 0; p < SPLITS; ++p)
      s += kvp[((size_t)bh * SPLITS + p) * F_ * D_ + fe];
    kv[i] = s;
  }
  if (i < BH * F_) {
    int bh = i / F_;
    int f = i % F_;
    float s = 0.0f;
#pragma unroll
    for (int p = 0; p < SPLITS; ++p)
      s += ksump[((size_t)bh * SPLITS + p) * F_ + f];
    ksum[i] = s;
  }
}

// ---------------------------------------------------------------------------
// Pass 2: q -> q' (in LDS) -> denom, out = (q' @ kv) / denom.
// grid = BH * (N/64) blocks, 512 threads (16 waves); 64 q-rows per block.
// ---------------------------------------------------------------------------
__global__ __launch_bounds__(512)
void FastAttention_out_kernel(const float* __restrict__ q,
                              const float* __restrict__ P,
                              const float* __restrict__ kv,
                              const float* __restrict__ ksum,
                              float* __restrict__ out)
{
  extern __shared__ float sm[];
  float* qpLDS  = sm;                          // 64 * 260
  float* kvLDS  = qpLDS + 64 * KP_PITCH;       // 256*64, layout [et][f][16]
  float* qLDS   = kvLDS + F_ * D_;             // 64 * 68
  float* ksLDS  = qLDS + 64 * KQ_PITCH;        // 256
  float* maxbuf = ksLDS + F_;                  // 64 * 16
  float* rmLDS  = maxbuf + 64 * 16;            // 64
  float* zLDS   = rmLDS + 64;                  // 64

  const int tid  = threadIdx.x;
  const int lane = tid & 31;
  const int wave = tid >> 5;
  const int l16  = lane & 15;
  const int hi   = lane >> 4;
  const int bh  = blockIdx.x >> 6;             // 64 blocks per (b,h)
  const int n0  = (blockIdx.x & 63) * 64;
  const float* qb  = q  + (size_t)bh * N_ * D_;
  const float* kvb = kv + (size_t)bh * F_ * D_;
  const int f0 = wave * 16;

  // Stage q tile (64x64), ksum, and kv (rearranged for e-tile reads).
  for (int i = tid; i < 64 * 64; i += 512)
    qLDS[(i >> 6) * KQ_PITCH + (i & 63)] =
        qb[(size_t)(n0 + (i >> 6)) * D_ + (i & 63)];
  for (int i = tid; i < F_; i += 512)
    ksLDS[i] = ksum[(size_t)bh * F_ + i] + 1e-6f;
  for (int i = tid; i < F_ * D_; i += 512) {
    int f = i >> 6, e = i & 63;
    kvLDS[(e >> 4) * (F_ * 16) + f * 16 + (e & 15)] = kvb[i];
  }

  // Preload projection B-fragments.
  v2f pB[16];
#pragma unroll
  for (int kk = 0; kk < 16; ++kk) {
    int d = kk * 4 + 2 * hi;
    pB[kk].x = P[(size_t)d * F_ + f0 + l16];
    pB[kk].y = P[(size_t)(d + 1) * F_ + f0 + l16];
  }
  __syncthreads();

  // Project all 4 row-chunks for this wave's feature block.
  v8f pr[4];
#pragma unroll
  for (int c = 0; c < 4; ++c) {
    v8f acc = {};
#pragma unroll
    for (int kk = 0; kk < 16; ++kk) {
      v2f a = *(const v2f*)&qLDS[(c * 16 + l16) * KQ_PITCH + kk * 4 + 2 * hi];
      acc = WMMA_F32(a, pB[kk], acc);
    }
    pr[c] = acc;
#pragma unroll
    for (int r = 0; r < 8; ++r) {
      float m = acc[r];
      m = fmaxf(m, __shfl_xor(m, 1, 32));
      m = fmaxf(m, __shfl_xor(m, 2, 32));
      m = fmaxf(m, __shfl_xor(m, 4, 32));
      m = fmaxf(m, __shfl_xor(m, 8, 32));
      if (l16 == 0) maxbuf[(c * 16 + r + hi * 8) * 16 + wave] = m;
    }
  }
  __syncthreads();
  if (tid < 64) {
    float m = maxbuf[tid * 16];
#pragma unroll
    for (int w = 1; w < 16; ++w) m = fmaxf(m, maxbuf[tid * 16 + w]);
    rmLDS[tid] = m;
  }
  __syncthreads();
#pragma unroll
  for (int c = 0; c < 4; ++c)
#pragma unroll
    for (int r = 0; r < 8; ++r) {
      float e = __expf(pr[c][r] - rmLDS[c * 16 + r + hi * 8]) * 0.0625f;
      qpLDS[(c * 16 + r + hi * 8) * KP_PITCH + f0 + l16] = e;
    }
  __syncthreads();

  // denom[row] = q'[row,:] . (ksum + 1e-6); z = 1/denom.
  if (tid < 64) {
    float d = 0.0f;
    for (int f = 0; f < F_; ++f) d += qpLDS[tid * KP_PITCH + f] * ksLDS[f];
    zLDS[tid] = 1.0f / d;
  }
  __syncthreads();

  // out tile: wave -> (row-chunk, e-tile); K = F = 256 (64 WMMA ops).
  const int c  = wave >> 2;
  const int et = wave & 3;
  v8f acc = {};
#pragma unroll
  for (int kk = 0; kk < 64; ++kk) {
    int f = kk * 4 + 2 * hi;
    v2f a = *(const v2f*)&qpLDS[(c * 16 + l16) * KP_PITCH + f];
    v2f b;
    b.x = kvLDS[et * (F_ * 16) + f * 16 + l16];
    b.y = kvLDS[et * (F_ * 16) + (f + 1) * 16 + l16];
    acc = WMMA_F32(a, b, acc);
  }

  float* ob = out + ((size_t)bh * N_ + n0) * D_;
#pragma unroll
  for (int r = 0; r < 8; ++r) {
    int row = c * 16 + r + hi * 8;
    ob[(size_t)row * D_ + et * 16 + l16] = acc[r] * zLDS[row];
  }
}

// ---------------------------------------------------------------------------
extern "C" void kernel_launch(void* const* d_in, const int* in_sizes, int n_in,
                              void* d_out, int out_size, void* d_ws, size_t ws_size,
                              hipStream_t stream) {
  const float* q = (const float*)d_in[0];
  const float* k = (const float*)d_in[1];
  const float* v = (const float*)d_in[2];
  const float* P = (const float*)d_in[3];
  float* out = (float*)d_out;

  float* ws    = (float*)d_ws;
  float* kvp   = ws;                                     // BH*SPLITS*F*D
  float* ksump = kvp + (size_t)BH * SPLITS * F_ * D_;    // BH*SPLITS*F
  float* kv    = ksump + (size_t)BH * SPLITS * F_;       // BH*F*D
  float* ksum  = kv + (size_t)BH * F_ * D_;              // BH*F

  FastAttention_kv_kernel<<<BH * SPLITS, 512, 0, stream>>>(k, v, P, kvp, ksump);
  FastAttention_reduce_kernel<<<(BH * F_ * D_ + 255) / 256, 256, 0, stream>>>(
      kvp, ksump, kv, ksum);

  size_t smem3 = (size_t)(64 * KP_PITCH + F_ * D_ + 64 * KQ_PITCH + F_ +
                          64 * 16 + 64 + 64) * sizeof(float);
  FastAttention_out_kernel<<<BH * (N_ / 64), 512, smem3, stream>>>(
      q, P, kv, ksum, out);
}